// LSTMDecoder_38714835206615
// MI455X (gfx1250) — compile-verified
//
#include <hip/hip_runtime.h>
#include <math.h>

typedef float v2f __attribute__((ext_vector_type(2)));
typedef float v8f __attribute__((ext_vector_type(8)));

#define SEQ    512
#define BATCH  128
#define D_EMB  512
#define D_HID  512
#define NG     2048            // 4*D_EMB (gates width)
#define GIN    1024            // D_EMB + D_HID
#define TSTEPS 511
#define MROWS  (TSTEPS * BATCH)  // 65408 rows of the big GEMM

#define LDSPITCH1 1028         // 1024 + 4 pad floats (bank-conflict dodge)
#define LDSPITCH2 516          // 512 + 4 pad floats
#define NBLK      32           // persistent LSTM grid size

// ---------------------------------------------------------------------------
// depthwise grouped conv + relu + pixel-shuffle-style upsample
// ---------------------------------------------------------------------------
__global__ __launch_bounds__(256) void upsample0_kernel(
    const float* __restrict__ z, const float* __restrict__ w,
    const float* __restrict__ b, float* __restrict__ z1) {
  int idx = blockIdx.x * 256 + threadIdx.x;     // flat over [256,128,512]
  if (idx >= 256 * 128 * 512) return;
  int c  = idx & 511;
  int bb = (idx >> 9) & 127;
  int t  = idx >> 16;
  int l  = t >> 1;
  int o  = ((t & 1) << 9) + c;                  // conv out channel 0..1023
  int cin = o >> 2;                             // groups=256 -> 4 out/group
  const float* wr = w + o * 7;
  float acc = b[o];
#pragma unroll
  for (int kk = 0; kk < 7; ++kk) {
    int li = l + kk - 3;
    if (li >= 0 && li < 128)
      acc = fmaf(wr[kk], z[((size_t)li * 128 + bb) * 256 + cin], acc);
  }
  z1[idx] = fmaxf(acc, 0.f);
}

__global__ __launch_bounds__(256) void upsample1_kernel(
    const float* __restrict__ z1, const float* __restrict__ w,
    const float* __restrict__ b, float* __restrict__ z2) {
  int idx = blockIdx.x * 256 + threadIdx.x;     // flat over [512,128,512]
  if (idx >= 512 * 128 * 512) return;
  int c  = idx & 511;
  int bb = (idx >> 9) & 127;
  int t  = idx >> 16;
  int l  = t >> 1;
  int o  = ((t & 1) << 9) + c;
  int cin = o >> 1;                             // groups=512 -> 2 out/group
  const float* wr = w + o * 7;
  float acc = b[o];
#pragma unroll
  for (int kk = 0; kk < 7; ++kk) {
    int li = l + kk - 3;
    if (li >= 0 && li < 256)
      acc = fmaf(wr[kk], z1[((size_t)li * 128 + bb) * 512 + cin], acc);
  }
  z2[idx] = fmaxf(acc, 0.f);
}

// ---------------------------------------------------------------------------
// gates_x = concat(x[:-1], z2[1:]) @ w_ih^T + b_ih + b_hh
// block = 128M x 64N tile, 8 waves; w_ih strip (64x1024 f32 = 257KB) in LDS
// ---------------------------------------------------------------------------
__global__ __launch_bounds__(256) void gatesx_wmma_kernel(
    const float* __restrict__ x, const float* __restrict__ z2,
    const float* __restrict__ w_ih, const float* __restrict__ b_ih,
    const float* __restrict__ b_hh, float* __restrict__ gates_x) {
  extern __shared__ float lds[];
  const int tid = threadIdx.x;
  const int mb  = blockIdx.x;      // 0..510  (128 rows each)
  const int nb  = blockIdx.y;      // 0..31   (64 gate cols each)

  // stage w_ih rows [64*nb, 64*nb+64) x K=1024 into LDS
  for (int i = tid; i < 64 * 256; i += 256) {
    int n = i >> 8, kq = i & 255;
    const float4 v = ((const float4*)(w_ih + (size_t)(nb * 64 + n) * GIN))[kq];
    *(float4*)(lds + n * LDSPITCH1 + kq * 4) = v;
  }
  __syncthreads();

  const int lane = tid & 31, wv = tid >> 5;
  const int lm = lane & 15, half = lane >> 4;
  const int koff = 2 * half;                     // ISA: lanes16-31 hold K=2,3
  const int Mtile = mb * 128 + wv * 16;

  const float* arow_x = x  + (size_t)(Mtile + lm) * D_EMB;          // k<512
  const float* arow_z = z2 + (size_t)(Mtile + lm + BATCH) * D_HID;  // k>=512 (t+1)
  const float* b0p = lds + (lm)      * (size_t)LDSPITCH1;
  const float* b1p = lds + (16 + lm) * (size_t)LDSPITCH1;
  const float* b2p = lds + (32 + lm) * (size_t)LDSPITCH1;
  const float* b3p = lds + (48 + lm) * (size_t)LDSPITCH1;

  v8f acc0 = {}, acc1 = {}, acc2 = {}, acc3 = {};
  for (int kc = 0; kc < 128; ++kc) {             // K = 0..511  (A from x)
    const int k = kc * 4 + koff;
    v2f a  = *(const v2f*)(arow_x + k);
    v2f b0 = *(const v2f*)(b0p + k);
    v2f b1 = *(const v2f*)(b1p + k);
    v2f b2 = *(const v2f*)(b2p + k);
    v2f b3 = *(const v2f*)(b3p + k);
    acc0 = __builtin_amdgcn_wmma_f32_16x16x4_f32(false, a, false, b0, (short)0, acc0, false, false);
    acc1 = __builtin_amdgcn_wmma_f32_16x16x4_f32(false, a, false, b1, (short)0, acc1, false, false);
    acc2 = __builtin_amdgcn_wmma_f32_16x16x4_f32(false, a, false, b2, (short)0, acc2, false, false);
    acc3 = __builtin_amdgcn_wmma_f32_16x16x4_f32(false, a, false, b3, (short)0, acc3, false, false);
  }
  for (int kc = 0; kc < 128; ++kc) {             // K = 512..1023 (A from z2)
    const int k  = kc * 4 + koff;
    const int kl = 512 + k;
    v2f a  = *(const v2f*)(arow_z + k);
    v2f b0 = *(const v2f*)(b0p + kl);
    v2f b1 = *(const v2f*)(b1p + kl);
    v2f b2 = *(const v2f*)(b2p + kl);
    v2f b3 = *(const v2f*)(b3p + kl);
    acc0 = __builtin_amdgcn_wmma_f32_16x16x4_f32(false, a, false, b0, (short)0, acc0, false, false);
    acc1 = __builtin_amdgcn_wmma_f32_16x16x4_f32(false, a, false, b1, (short)0, acc1, false, false);
    acc2 = __builtin_amdgcn_wmma_f32_16x16x4_f32(false, a, false, b2, (short)0, acc2, false, false);
    acc3 = __builtin_amdgcn_wmma_f32_16x16x4_f32(false, a, false, b3, (short)0, acc3, false, false);
  }

  const int n = nb * 64 + lm;
  const float bi0 = b_ih[n]      + b_hh[n];
  const float bi1 = b_ih[n + 16] + b_hh[n + 16];
  const float bi2 = b_ih[n + 32] + b_hh[n + 32];
  const float bi3 = b_ih[n + 48] + b_hh[n + 48];
#pragma unroll
  for (int vc = 0; vc < 8; ++vc) {               // C/D: vgpr j -> rows j, j+8
    const size_t row = (size_t)(Mtile + vc + 8 * half) * NG;
    gates_x[row + n]      = acc0[vc] + bi0;
    gates_x[row + n + 16] = acc1[vc] + bi1;
    gates_x[row + n + 32] = acc2[vc] + bi2;
    gates_x[row + n + 48] = acc3[vc] + bi3;
  }
}

// ---------------------------------------------------------------------------
// persistent LSTM scan, gate-interleaved column ownership:
//   block b owns j in [16b,16b+16); its 4 WMMA tiles per wave are the
//   i/f/g/o quadrant columns {512q + 16b + [0,16)}, so the pointwise LSTM
//   update is done entirely in registers (c never leaves VGPRs).
//   h is exchanged through a global ping-pong pair -> ONE grid sync per step.
// ---------------------------------------------------------------------------
__device__ __forceinline__ float sigmoidf_(float v) { return 1.0f / (1.0f + expf(-v)); }

__device__ __forceinline__ void grid_sync(unsigned* cnt, volatile unsigned* gen) {
  __threadfence();
  __syncthreads();
  if (threadIdx.x == 0) {
    const unsigned my = *gen;
    const unsigned prev = atomicAdd(cnt, 1u);
    if (prev == NBLK - 1) {
      atomicExch(cnt, 0u);
      __threadfence();
      atomicAdd((unsigned*)gen, 1u);
    } else {
      while (*gen == my) { __builtin_amdgcn_s_sleep(1); }
    }
  }
  __syncthreads();
  __threadfence();
}

__global__ __launch_bounds__(256) void lstm_scan_kernel(
    const float* __restrict__ gates_x, const float* __restrict__ w_hh,
    float* __restrict__ h0, float* __restrict__ h1,
    float* __restrict__ out, unsigned* __restrict__ bar) {
  extern __shared__ float lds[];
  const int tid = threadIdx.x;
  const int nb  = blockIdx.x;                    // j range [16*nb, 16*nb+16)

  // stage w_hh rows {512q + 16nb + i : q=0..3, i=0..15} x K=512 (129 KB)
  for (int idx = tid; idx < 64 * 128; idx += 256) {
    int n = idx >> 7, kq = idx & 127;
    int q = n >> 4, i = n & 15;
    const size_t grow = (size_t)(q * 512 + nb * 16 + i);
    const float4 v = ((const float4*)(w_hh + grow * D_EMB))[kq];
    *(float4*)(lds + n * LDSPITCH2 + kq * 4) = v;
  }
  __syncthreads();

  const int lane = tid & 31, wv = tid >> 5;
  const int lm = lane & 15, half = lane >> 4;
  const int koff = 2 * half;
  const float* b0p = lds + (lm)      * (size_t)LDSPITCH2;   // i quadrant
  const float* b1p = lds + (16 + lm) * (size_t)LDSPITCH2;   // f quadrant
  const float* b2p = lds + (32 + lm) * (size_t)LDSPITCH2;   // g quadrant
  const float* b3p = lds + (48 + lm) * (size_t)LDSPITCH2;   // o quadrant
  const int j = nb * 16 + lm;                    // this lane's hidden column
  unsigned* cnt = bar;
  volatile unsigned* gen = bar + 1;

  float creg[8] = {0.f, 0.f, 0.f, 0.f, 0.f, 0.f, 0.f, 0.f};  // cell state

  for (int t = 0; t < TSTEPS; ++t) {
    const float* hin  = (t & 1) ? h1 : h0;
    float*       hout = (t & 1) ? h0 : h1;

    const float* hrow = hin + (size_t)(wv * 16 + lm) * D_EMB;
    v8f acc0 = {}, acc1 = {}, acc2 = {}, acc3 = {};
    for (int kc = 0; kc < 128; ++kc) {
      const int k = kc * 4 + koff;
      v2f a  = *(const v2f*)(hrow + k);
      v2f b0 = *(const v2f*)(b0p + k);
      v2f b1 = *(const v2f*)(b1p + k);
      v2f b2 = *(const v2f*)(b2p + k);
      v2f b3 = *(const v2f*)(b3p + k);
      acc0 = __builtin_amdgcn_wmma_f32_16x16x4_f32(false, a, false, b0, (short)0, acc0, false, false);
      acc1 = __builtin_amdgcn_wmma_f32_16x16x4_f32(false, a, false, b1, (short)0, acc1, false, false);
      acc2 = __builtin_amdgcn_wmma_f32_16x16x4_f32(false, a, false, b2, (short)0, acc2, false, false);
      acc3 = __builtin_amdgcn_wmma_f32_16x16x4_f32(false, a, false, b3, (short)0, acc3, false, false);
    }

    // fused gate math in registers; acc0..3 are i,f,g,o for (m, j)
    const float* gx = gates_x + (size_t)t * (BATCH * NG);
#pragma unroll
    for (int vc = 0; vc < 8; ++vc) {
      const int m = wv * 16 + vc + 8 * half;
      const size_t row = (size_t)m * NG;
      const float ig = acc0[vc] + gx[row + j];
      const float fg = acc1[vc] + gx[row + 512 + j];
      const float gg = acc2[vc] + gx[row + 1024 + j];
      const float og = acc3[vc] + gx[row + 1536 + j];
      const float cc = sigmoidf_(fg) * creg[vc] + sigmoidf_(ig) * tanhf(gg);
      const float hh = sigmoidf_(og) * tanhf(cc);
      creg[vc] = cc;
      hout[(size_t)m * D_EMB + j] = hh;
      out[(size_t)t * (BATCH * D_EMB) + (size_t)m * D_EMB + j] = hh;
    }
    grid_sync(cnt, gen);     // h ping-pong: one barrier per timestep
  }
}

// ---------------------------------------------------------------------------
extern "C" void kernel_launch(void* const* d_in, const int* in_sizes, int n_in,
                              void* d_out, int out_size, void* d_ws, size_t ws_size,
                              hipStream_t stream) {
  const float* x       = (const float*)d_in[0];
  const float* z       = (const float*)d_in[1];
  const float* conv0_w = (const float*)d_in[2];
  const float* conv0_b = (const float*)d_in[3];
  const float* conv1_w = (const float*)d_in[4];
  const float* conv1_b = (const float*)d_in[5];
  const float* w_ih    = (const float*)d_in[6];
  const float* w_hh    = (const float*)d_in[7];
  const float* b_ih    = (const float*)d_in[8];
  const float* b_hh    = (const float*)d_in[9];
  float* out = (float*)d_out;

  float* ws = (float*)d_ws;
  size_t off = 0;
  float* z1     = ws + off; off += (size_t)256 * 128 * 512;
  float* z2     = ws + off; off += (size_t)512 * 128 * 512;
  float* gatesx = ws + off; off += (size_t)MROWS * NG;
  float* h0     = ws + off; off += (size_t)BATCH * D_EMB;
  float* h1     = ws + off; off += (size_t)BATCH * D_EMB;
  unsigned* bar = (unsigned*)(ws + off);

  hipMemsetAsync(h0, 0, BATCH * D_EMB * sizeof(float), stream);  // h(-1) = 0
  hipMemsetAsync(bar, 0, 2 * sizeof(unsigned), stream);

  upsample0_kernel<<<(256 * 128 * 512) / 256, 256, 0, stream>>>(z, conv0_w, conv0_b, z1);
  upsample1_kernel<<<(512 * 128 * 512) / 256, 256, 0, stream>>>(z1, conv1_w, conv1_b, z2);
  gatesx_wmma_kernel<<<dim3(511, 32), 256, 64 * LDSPITCH1 * sizeof(float), stream>>>(
      x, z2, w_ih, b_ih, b_hh, gatesx);
  lstm_scan_kernel<<<NBLK, 256, 64 * LDSPITCH2 * sizeof(float), stream>>>(
      gatesx, w_hh, h0, h1, out, bar);
}